// AllToAllDispatchForward_15822659519276
// MI455X (gfx1250) — compile-verified
//
#include <hip/hip_runtime.h>

// MoE all-to-all token dispatch for MI455X (gfx1250).
// out[d][slot][:] = x[slot/K][:] if expert_mapping[expert_indices[slot]]==d else 0
// Streaming problem: 512 MiB stores + ~32 MiB loads => ~24.5 us floor @ 23.3 TB/s.
// Straight-line kernel: one block per (slot, H-chunk); async global->LDS b128
// stage (ASYNCcnt), then 8 branchless non-temporal b128 stores across planes.

typedef float v4f __attribute__((ext_vector_type(4)));

#define NUM_DEVS 8
#define BLOCK    128
#define CHUNK    (BLOCK * 4) // 512 floats per block == H for the reference shapes

#if defined(__gfx1250__) &&                                              \
    __has_builtin(__builtin_amdgcn_global_load_async_to_lds_b128) &&     \
    __has_builtin(__builtin_amdgcn_s_wait_asynccnt)
#define USE_ASYNC_LDS 1
#else
#define USE_ASYNC_LDS 0
#endif

#if USE_ASYNC_LDS
// Parameter type per hipcc diagnostic: pointer to int4 (vector_size(16)) in
// AS1 (global) / AS3 (LDS).
typedef int async_v4i __attribute__((vector_size(4 * sizeof(int))));

__device__ __forceinline__ void async_copy_b128(const float* gsrc, float* lds_dst) {
    __builtin_amdgcn_global_load_async_to_lds_b128(
        (__attribute__((address_space(1))) async_v4i*)gsrc,
        (__attribute__((address_space(3))) async_v4i*)lds_dst,
        0, 0);
}
#endif

// KK > 0: compile-time top-k (unsigned divide folds to a shift).
// KK == 0: runtime k fallback.
template <unsigned KK>
__global__ __launch_bounds__(BLOCK) void moe_a2a_dispatch(
    const float* __restrict__ x,     // [T, H]
    const int*   __restrict__ eidx,  // [T*K]
    const int*   __restrict__ emap,  // [num_experts]
    float*       __restrict__ out,   // [D, T*K, H]
    int H, unsigned k_rt, size_t rowStride /* = T*K*H, one device plane */)
{
#if USE_ASYNC_LDS
    __shared__ float stage[CHUNK];
#endif
    const unsigned slot = blockIdx.x;                      // flat (token, k)
    const unsigned t    = (KK > 0) ? (slot / KK) : (slot / k_rt);
    const unsigned tid  = threadIdx.x;
    const int h         = (int)(blockIdx.y * CHUNK + tid * 4); // gridDim.y = ceil(H/CHUNK)

    if (h < H) {
        // Block-uniform routing lookup (scalar loads); strictly 32-bit so the
        // per-plane compares stay s_cmp_eq_i32.
        const int mydev = emap[eidx[slot]];

        const float* gsrc = x + (size_t)t * (size_t)H + (size_t)h;

#if USE_ASYNC_LDS
        // CDNA5 async copy (ASYNCcnt): global -> LDS, each lane stages its own
        // 16 bytes; latency hidden by the ~131K-wave oversubscription.
        async_copy_b128(gsrc, &stage[tid * 4]);
        __builtin_amdgcn_s_wait_asynccnt(0);
        const v4f data = *(const v4f*)&stage[tid * 4];
#else
        const v4f data = *(const v4f*)gsrc;
#endif

        // Branchless: walk the 8 device planes, storing data on the owning
        // plane and zeros elsewhere. Non-temporal b128: the 512 MB streamed
        // output stays out of L2 so the K-times-reused token tensor stays hot.
        const v4f zero4 = (v4f)0.0f;
        float* dst = out + (size_t)slot * (size_t)H + (size_t)h;
#pragma unroll
        for (int d = 0; d < NUM_DEVS; ++d) {
            const v4f v = (d == mydev) ? data : zero4;
            __builtin_nontemporal_store(v, (v4f*)dst);
            dst += rowStride;
        }
    }
}

extern "C" void kernel_launch(void* const* d_in, const int* in_sizes, int n_in,
                              void* d_out, int out_size, void* d_ws, size_t ws_size,
                              hipStream_t stream) {
    (void)n_in; (void)d_ws; (void)ws_size;
    const float* x    = (const float*)d_in[0];   // [B,S,H] f32
    const int*   eidx = (const int*)d_in[1];     // [B,S,K] i32
    const int*   emap = (const int*)d_in[2];     // [E] i32
    float*       out  = (float*)d_out;           // [D, T*K, H] f32

    // Derive shapes: in_sizes[0]=T*H, in_sizes[1]=T*K, out_size=D*T*K*H.
    const long long D  = NUM_DEVS;
    const long long TK = (long long)in_sizes[1];
    const int H = (int)(((long long)out_size / D) / TK);
    const int T = (int)((long long)in_sizes[0] / H);
    const unsigned K = (unsigned)(TK / T);
    const size_t rowStride = (size_t)TK * (size_t)H;

    const dim3 grid((unsigned)TK, (unsigned)((H + CHUNK - 1) / CHUNK));
    const dim3 block(BLOCK);

    if (K == 2) {
        moe_a2a_dispatch<2u><<<grid, block, 0, stream>>>(x, eidx, emap, out, H, K, rowStride);
    } else {
        moe_a2a_dispatch<0u><<<grid, block, 0, stream>>>(x, eidx, emap, out, H, K, rowStride);
    }
}